// Net_72877005078649
// MI455X (gfx1250) — compile-verified
//
#include <hip/hip_runtime.h>
#include <hip/hip_bf16.h>

#define N_TRAIN 4096
#define IN_DIM  64
#define HID     128
#define OUT_C   10

typedef float v2f __attribute__((ext_vector_type(2)));
typedef float v8f __attribute__((ext_vector_type(8)));

// --------------------------------------------------------------------------
// Kernel 1: fused projection  P = relu(X * W1^T) * W2^T   using
// V_WMMA_F32_16X16X4_F32 (fp32 matrix cores, CDNA5).
// One block (256 thr = 8 waves) per 16-row M-tile; grid.y: 0 -> x, 1 -> train_X.
//
// Fragment layouts (ISA 7.12.2, wave32):
//   A 16x4 f32 : lane L<16 holds row M=L, K={0,1}; lane L>=16 holds M=L-16, K={2,3}
//   B 4x16 f32 : lane L<16 holds col N=L, K={0,1}; lane L>=16 holds N=L-16, K={2,3}
//   C/D 16x16  : VGPR v, lanes 0-15 -> (M=v, N=lane); lanes 16-31 -> (M=v+8, N=lane-16)
// --------------------------------------------------------------------------
__global__ __launch_bounds__(256)
void proj_wmma_kernel(const float* __restrict__ X0,
                      const float* __restrict__ X1,
                      const float* __restrict__ W1,
                      const float* __restrict__ W2,
                      float* __restrict__ ws)
{
    const float* X = (blockIdx.y == 0) ? X0 : X1;
    float* P = ws + (size_t)blockIdx.y * ((size_t)N_TRAIN * OUT_C);

    const int tid  = threadIdx.x;
    const int wave = tid >> 5;
    const int lane = tid & 31;
    const int half = lane >> 4;   // 0: K pair {0,1}, 1: K pair {2,3}
    const int mr   = lane & 15;   // A row / B col / C col index

    const int Mbase = blockIdx.x * 16;
    const int Nbase = wave * 16;  // HID tile handled by this wave

    __shared__ float Hs[16][HID + 4];   // +4 padding to break bank conflicts

    // ---- GEMM1: H_tile(16x16) = X_tile(16x64) x W1^T(64x16) ----
    const float* Arow = X  + (size_t)(Mbase + mr) * IN_DIM;
    const float* Brow = W1 + (size_t)(Nbase + mr) * IN_DIM;   // W1 is [HID][IN_DIM]

    v8f acc = {};
    #pragma unroll
    for (int k0 = 0; k0 < IN_DIM; k0 += 4) {
        v2f a = *(const v2f*)(Arow + k0 + 2 * half);
        v2f b = *(const v2f*)(Brow + k0 + 2 * half);
        acc = __builtin_amdgcn_wmma_f32_16x16x4_f32(false, a, false, b,
                                                    (short)0, acc, false, false);
    }
    // relu + stage to LDS
    #pragma unroll
    for (int v = 0; v < 8; ++v) {
        float r = acc[v];
        Hs[v + 8 * half][Nbase + mr] = (r > 0.f) ? r : 0.f;
    }
    __syncthreads();

    // ---- GEMM2 (wave 0): P_tile(16 x 16pad) = H(16x128) x W2^T(128x16pad) ----
    if (wave == 0) {
        const float* w2row = W2 + (size_t)mr * HID;  // W2 is [OUT][HID]; valid mr<10
        v8f acc2 = {};
        #pragma unroll
        for (int k0 = 0; k0 < HID; k0 += 4) {
            v2f a;
            a.x = Hs[mr][k0 + 2 * half];
            a.y = Hs[mr][k0 + 2 * half + 1];
            v2f b;
            if (mr < OUT_C) b = *(const v2f*)(w2row + k0 + 2 * half);
            else            { b.x = 0.f; b.y = 0.f; }
            acc2 = __builtin_amdgcn_wmma_f32_16x16x4_f32(false, a, false, b,
                                                         (short)0, acc2, false, false);
        }
        if (mr < OUT_C) {
            #pragma unroll
            for (int v = 0; v < 8; ++v)
                P[(size_t)(Mbase + v + 8 * half) * OUT_C + mr] = acc2[v];
        }
    }
}

// --------------------------------------------------------------------------
// Kernel 2: leave-one-out Nadaraya-Watson regression.
// One block per query row b; threads stride over the 4096 reference rows.
// Transcendental-bound: one v_exp_f32 per (b,n,o) term via exp2.
// --------------------------------------------------------------------------
__global__ __launch_bounds__(256)
void nw_loo_kernel(const float* __restrict__ Zw,
                   const float* __restrict__ Xw,
                   const float* __restrict__ Y,
                   const float* __restrict__ hp,
                   float* __restrict__ out)
{
    const int b    = blockIdx.x;
    const int tid  = threadIdx.x;
    const int wave = tid >> 5;
    const int lane = tid & 31;

    __shared__ float Zs[OUT_C];
    __shared__ float red[8][2 * OUT_C];

    if (tid < OUT_C) Zs[tid] = Zw[(size_t)b * OUT_C + tid];
    __syncthreads();

    const float inv_h = 1.0f / hp[0];
    const float cexp  = -0.5f * 1.44269504088896340736f;   // -0.5 * log2(e)

    float zb[OUT_C];
    #pragma unroll
    for (int o = 0; o < OUT_C; ++o) zb[o] = Zs[o];

    float num[OUT_C], den[OUT_C];
    #pragma unroll
    for (int o = 0; o < OUT_C; ++o) { num[o] = 0.f; den[o] = 0.f; }

    for (int n = tid; n < N_TRAIN; n += 256) {
        if (n + 256 < N_TRAIN) {
            __builtin_prefetch(Xw + (size_t)(n + 256) * OUT_C, 0, 3);
            __builtin_prefetch(Y  + (size_t)(n + 256) * OUT_C, 0, 3);
        }
        if (n == b) continue;   // leave-one-out: zero the j==i term
        const float* xr = Xw + (size_t)n * OUT_C;
        const float* yr = Y  + (size_t)n * OUT_C;
        #pragma unroll
        for (int o = 0; o < OUT_C; ++o) {
            float d = (xr[o] - zb[o]) * inv_h;
            float w = __builtin_amdgcn_exp2f(cexp * d * d);
            num[o] += w * yr[o];
            den[o] += w;
        }
    }

    // deterministic intra-wave reduction (wave32)
    #pragma unroll
    for (int off = 16; off > 0; off >>= 1) {
        #pragma unroll
        for (int o = 0; o < OUT_C; ++o) {
            num[o] += __shfl_down(num[o], off, 32);
            den[o] += __shfl_down(den[o], off, 32);
        }
    }
    if (lane == 0) {
        #pragma unroll
        for (int o = 0; o < OUT_C; ++o) {
            red[wave][o]         = num[o];
            red[wave][OUT_C + o] = den[o];
        }
    }
    __syncthreads();

    if (tid < OUT_C) {
        float nn = 0.f, dd = 0.f;
        #pragma unroll
        for (int w = 0; w < 8; ++w) { nn += red[w][tid]; dd += red[w][OUT_C + tid]; }
        out[(size_t)b * OUT_C + tid] = nn / dd;
    }
}

// --------------------------------------------------------------------------
extern "C" void kernel_launch(void* const* d_in, const int* in_sizes, int n_in,
                              void* d_out, int out_size, void* d_ws, size_t ws_size,
                              hipStream_t stream) {
    const float* x  = (const float*)d_in[0];
    const float* tX = (const float*)d_in[1];
    const float* Y  = (const float*)d_in[2];
    const float* W1 = (const float*)d_in[3];
    const float* W2 = (const float*)d_in[4];
    const float* h  = (const float*)d_in[5];
    float* out = (float*)d_out;
    float* ws  = (float*)d_ws;   // layout: Zw[4096*10] | Xw[4096*10]  (320 KB)

    dim3 g1(N_TRAIN / 16, 2);
    proj_wmma_kernel<<<g1, 256, 0, stream>>>(x, tX, W1, W2, ws);

    const float* Zw = ws;
    const float* Xw = ws + (size_t)N_TRAIN * OUT_C;
    nw_loo_kernel<<<N_TRAIN, 256, 0, stream>>>(Zw, Xw, Y, h, out);

    (void)in_sizes; (void)n_in; (void)out_size; (void)ws_size;
}